// LightGCN_44581760532488
// MI455X (gfx1250) — compile-verified
//
#include <hip/hip_runtime.h>
#include <stdint.h>

#define N_USERS 50000
#define N_ITEMS 50000
#define N_NODES (N_USERS + N_ITEMS)
#define D       256
#define N_EDGES 6400000
#define N_LAYERS 5
#define BATCH   4096
#define CHUNK   256   // edges staged per block

typedef float v2f __attribute__((ext_vector_type(2)));
typedef float v8f __attribute__((ext_vector_type(8)));

// ---------- gfx1250 async-to-LDS helpers ----------
__device__ __forceinline__ uint32_t lds_addr32(const void* p) {
  // low 32 bits of a generic pointer into LDS == LDS byte offset (aperture rule)
  return (uint32_t)(uintptr_t)p;
}
__device__ __forceinline__ void async_b32_to_lds(uint32_t lds_off, const void* gaddr) {
  uint64_t ga = (uint64_t)(uintptr_t)gaddr;
  asm volatile("global_load_async_to_lds_b32 %0, %1, off"
               :: "v"(lds_off), "v"(ga)
               : "memory");
}
__device__ __forceinline__ void wait_asynccnt0() {
  asm volatile("s_wait_asynccnt 0" ::: "memory");
}

// ---------- build x0 = concat(user_table, item_table) ----------
__global__ void concat_kernel(float4* __restrict__ dst,
                              const float4* __restrict__ usr,
                              const float4* __restrict__ itm) {
  size_t idx = (size_t)blockIdx.x * blockDim.x + threadIdx.x;  // over N_NODES*D/4
  const size_t uN = (size_t)N_USERS * (D / 4);
  const size_t tN = (size_t)N_NODES * (D / 4);
  if (idx >= tN) return;
  dst[idx] = (idx < uN) ? usr[idx] : itm[idx - uN];
}

__global__ void zero_kernel(float4* __restrict__ p, size_t n4) {
  size_t idx = (size_t)blockIdx.x * blockDim.x + threadIdx.x;
  if (idx < n4) p[idx] = make_float4(0.f, 0.f, 0.f, 0.f);
}

// ---------- SpMM: y += vals[e] * x[col[e]]  scattered to row[e] ----------
// One block stages CHUNK edges of metadata into LDS via the async pipe,
// then 64 threads/edge each move one float4 (64*4 = 256 dims) with f32 atomics.
__global__ void spmm_kernel(float* __restrict__ y, const float* __restrict__ x,
                            const int* __restrict__ row, const int* __restrict__ col,
                            const float* __restrict__ vals, int nEdges) {
  __shared__ int   sh_row[CHUNK];
  __shared__ int   sh_col[CHUNK];
  __shared__ float sh_val[CHUNK];

  const int t = threadIdx.x;                       // 256 threads
  const long long base = (long long)blockIdx.x * CHUNK;

  // async stage of this block's edge metadata (exact grid: never OOB)
  async_b32_to_lds(lds_addr32(&sh_row[t]), row  + base + t);
  async_b32_to_lds(lds_addr32(&sh_col[t]), col  + base + t);
  async_b32_to_lds(lds_addr32(&sh_val[t]), vals + base + t);
  wait_asynccnt0();
  __syncthreads();

  const int sub  = t & 63;    // which float4 of the 256-dim row
  const int eloc = t >> 6;    // 0..3 : 4 edges in flight per iteration

  for (int i = 0; i < CHUNK / 4; ++i) {
    const int slot = i * 4 + eloc;
    const long long e = base + slot;
    if (e >= nEdges) break;
    const int   r = sh_row[slot];
    const int   c = sh_col[slot];
    const float v = sh_val[slot];
    const float4 xv = ((const float4*)(x + (size_t)c * D))[sub];
    float* yr = y + (size_t)r * D + (size_t)sub * 4;
    unsafeAtomicAdd(yr + 0, v * xv.x);
    unsafeAtomicAdd(yr + 1, v * xv.y);
    unsafeAtomicAdd(yr + 2, v * xv.z);
    unsafeAtomicAdd(yr + 3, v * xv.w);
  }
}

// ---------- WMMA epilogue: out[i][:] += (1/6) * x[node(i)][:] ----------
// Tile: 16 out-rows x 16 dims per wave. Per tile, 4 chained
// v_wmma_f32_16x16x4_f32 with A = (1/6)*selection (rows 4g..4g+3), B = the
// 4 gathered x rows, C accumulating the running out tile. Exact in f32.
__global__ void gather_accum_wmma(float* __restrict__ out, const float* __restrict__ x,
                                  const int* __restrict__ users, const int* __restrict__ items,
                                  int accumulate) {
  const int lane   = threadIdx.x & 31;
  const int waveId = threadIdx.x >> 5;             // 8 waves/block
  const int tile   = blockIdx.x * 8 + waveId;      // 0..8191
  const int tm     = tile >> 4;                    // row-tile 0..511
  const int tn     = tile & 15;                    // col-tile 0..15

  const int hi = lane >> 4;                        // lane half
  const int n  = lane & 15;                        // N / M index for this lane
  const int colBase = tn * 16;

  // C/D layout: VGPR j, lanes 0-15 -> (M=j, N=lane); lanes 16-31 -> (M=j+8, N=lane-16)
  v8f c;
#pragma unroll
  for (int j = 0; j < 8; ++j) {
    const int r = tm * 16 + j + hi * 8;
    c[j] = accumulate ? out[(size_t)r * D + colBase + n] : 0.0f;
  }

  const float s = 1.0f / 6.0f;
#pragma unroll
  for (int g = 0; g < 4; ++g) {
    const int kA = hi * 2;                         // K held in component 0 (K=0 or 2)
    const int kB = kA + 1;                         // K held in component 1 (K=1 or 3)
    const int rA = tm * 16 + g * 4 + kA;
    const int rB = tm * 16 + g * 4 + kB;
    const int nodeA = (rA < BATCH) ? users[rA] : (N_USERS + items[rA - BATCH]);
    const int nodeB = (rB < BATCH) ? users[rB] : (N_USERS + items[rB - BATCH]);
    // B (4x16): row K striped across lanes: VGPR0 = K0 (lanes0-15)/K2 (16-31), VGPR1 = K1/K3
    v2f b;
    b[0] = x[(size_t)nodeA * D + colBase + n];
    b[1] = x[(size_t)nodeB * D + colBase + n];
    // A (16x4): lane m = lane&15; VGPR0 = K (0|2), VGPR1 = K (1|3)
    v2f a;
    a[0] = ((lane & 15) == g * 4 + kA) ? s : 0.0f;
    a[1] = ((lane & 15) == g * 4 + kB) ? s : 0.0f;
    c = __builtin_amdgcn_wmma_f32_16x16x4_f32(false, a, false, b, (short)0, c,
                                              false, false);
  }

#pragma unroll
  for (int j = 0; j < 8; ++j) {
    const int r = tm * 16 + j + hi * 8;
    out[(size_t)r * D + colBase + n] = c[j];
  }
}

extern "C" void kernel_launch(void* const* d_in, const int* in_sizes, int n_in,
                              void* d_out, int out_size, void* d_ws, size_t ws_size,
                              hipStream_t stream) {
  (void)in_sizes; (void)n_in; (void)out_size; (void)ws_size;

  const float* user_table = (const float*)d_in[0];
  const float* item_table = (const float*)d_in[1];
  const float* adj_vals   = (const float*)d_in[2];
  const int*   adj_row    = (const int*)  d_in[3];
  const int*   adj_col    = (const int*)  d_in[4];
  const int*   users      = (const int*)  d_in[5];
  const int*   items      = (const int*)  d_in[6];
  float*       out        = (float*)d_out;

  float* xA = (float*)d_ws;
  float* xB = xA + (size_t)N_NODES * D;            // ping-pong, 102.4 MB each

  const size_t n4   = (size_t)N_NODES * D / 4;     // 6,400,000 float4
  const unsigned gN = (unsigned)((n4 + 255) / 256);// 25000 blocks

  // x0 = concat(user_table, item_table)
  concat_kernel<<<gN, 256, 0, stream>>>((float4*)xA, (const float4*)user_table,
                                        (const float4*)item_table);
  // out = x0[gather] / 6   (beta = 0: also initializes poisoned d_out)
  gather_accum_wmma<<<1024, 256, 0, stream>>>(out, xA, users, items, 0);

  for (int l = 0; l < N_LAYERS; ++l) {
    zero_kernel<<<gN, 256, 0, stream>>>((float4*)xB, n4);
    spmm_kernel<<<N_EDGES / CHUNK, 256, 0, stream>>>(xB, xA, adj_row, adj_col,
                                                     adj_vals, N_EDGES);
    gather_accum_wmma<<<1024, 256, 0, stream>>>(out, xB, users, items, 1);
    float* tmp = xA; xA = xB; xB = tmp;
  }
}